// GNN_GATConv_homogen_2422361555199
// MI455X (gfx1250) — compile-verified
//
#include <hip/hip_runtime.h>
#include <hip/hip_bf16.h>

// ---------------------------------------------------------------------------
// GAT link predictor for MI455X (gfx1250, wave32).
// GEMMs use V_WMMA_F32_16X16X4_F32 (fp32 in / fp32 acc) — exact-precision
// matrix path. Edge softmax/aggregation is L2-atomic bound; uses float4
// gathers + f32 global atomics, segment-max via order-preserving uint keys.
// ---------------------------------------------------------------------------

typedef __attribute__((ext_vector_type(2))) float v2f;
typedef __attribute__((ext_vector_type(8))) float v8f;

#define N_NODES   10000
#define FDIM      256
#define E_POS     320000
#define E_TOT     330000   // + self loops
#define E_PRED    100000
#define NEG_SLOPE 0.2f
#define EPS_F     1e-16f

// ---------------------------------------------------------------------------
// GEMM: Y[M,256] = X[M,256] @ W[256,256] (+ bias). One wave per 16x16 tile.
// M must be a multiple of 16 (10000 = 625*16). 64 chained K=4 WMMAs.
// ---------------------------------------------------------------------------
__global__ void wmma_gemm_f32(const float* __restrict__ X,
                              const float* __restrict__ W,
                              const float* __restrict__ bias,
                              float* __restrict__ Y, int M) {
  const int F = FDIM;
  const int wave = (blockIdx.x * blockDim.x + threadIdx.x) >> 5;
  const int lane = threadIdx.x & 31;
  const int ntiles = F / 16;                 // 16
  const int mtile = wave / ntiles;
  const int ntile = wave % ntiles;
  if (mtile * 16 >= M) return;               // wave-uniform

  const int half = lane >> 4;                // 0: lanes 0-15, 1: lanes 16-31
  const int lsub = lane & 15;

  const float* __restrict__ xrow = X + (size_t)(mtile * 16 + lsub) * F;
  const float* __restrict__ wcol = W + ntile * 16 + lsub;

  v8f c = {};
  #pragma unroll 8
  for (int k0 = 0; k0 < F; k0 += 4) {
    // A 16x4 f32 layout: lane(lsub)=row M, VGPR j = K = k0 + 2*half + j
    v2f a, b;
    a.x = xrow[k0 + 2 * half + 0];
    a.y = xrow[k0 + 2 * half + 1];
    // B 4x16 f32 layout: row K striped across lanes; lane half selects K pair
    b.x = wcol[(size_t)(k0 + 2 * half + 0) * F];
    b.y = wcol[(size_t)(k0 + 2 * half + 1) * F];
    c = __builtin_amdgcn_wmma_f32_16x16x4_f32(false, a, false, b,
                                              (short)0, c, false, false);
  }

  const float bv = bias ? bias[ntile * 16 + lsub] : 0.0f;
  #pragma unroll
  for (int r = 0; r < 8; ++r) {
    // C/D 16x16 f32 layout: VGPR r -> row M = r + 8*half, col N = lsub
    Y[(size_t)(mtile * 16 + r + 8 * half) * F + ntile * 16 + lsub] = c[r] + bv;
  }
}

// ---------------------------------------------------------------------------
// al_s[n] = h[n,:] . a_src ; al_d[n] = h[n,:] . a_dst  (wave per node)
// ---------------------------------------------------------------------------
__global__ void rowdot2_kernel(const float* __restrict__ h,
                               const float* __restrict__ as,
                               const float* __restrict__ ad,
                               float* __restrict__ als,
                               float* __restrict__ ald, int N) {
  const int wave = (blockIdx.x * blockDim.x + threadIdx.x) >> 5;
  const int lane = threadIdx.x & 31;
  if (wave >= N) return;
  const float* row = h + (size_t)wave * FDIM;
  float s0 = 0.0f, s1 = 0.0f;
  #pragma unroll
  for (int f = lane; f < FDIM; f += 32) {
    const float v = row[f];
    s0 += v * as[f];
    s1 += v * ad[f];
  }
  #pragma unroll
  for (int m = 16; m > 0; m >>= 1) {
    s0 += __shfl_xor(s0, m, 32);
    s1 += __shfl_xor(s1, m, 32);
  }
  if (lane == 0) { als[wave] = s0; ald[wave] = s1; }
}

// ---------------------------------------------------------------------------
// out[n,f] = bias[f]  (fold "+ bias" into the scatter accumulator init)
// ---------------------------------------------------------------------------
__global__ void init_out_bias(float* __restrict__ out,
                              const float* __restrict__ bias, int total) {
  const int i = blockIdx.x * blockDim.x + threadIdx.x;
  if (i < total) out[i] = bias[i & (FDIM - 1)];
}

__global__ void init_ms_kernel(unsigned* __restrict__ mkeys,
                               float* __restrict__ ssum, int N) {
  const int i = blockIdx.x * blockDim.x + threadIdx.x;
  if (i < N) { mkeys[i] = 0u; ssum[i] = 0.0f; }
}

// order-preserving float <-> uint mapping (monotone increasing)
__device__ __forceinline__ unsigned enc_f32(float f) {
  unsigned u = __float_as_uint(f);
  return (u & 0x80000000u) ? ~u : (u | 0x80000000u);
}
__device__ __forceinline__ float dec_f32(unsigned k) {
  return (k & 0x80000000u) ? __uint_as_float(k & 0x7fffffffu)
                           : __uint_as_float(~k);
}

__device__ __forceinline__ void edge_sd(const int* __restrict__ pei, int e,
                                        int& s, int& d) {
  if (e < E_POS) { s = pei[e]; d = pei[E_POS + e]; }
  else           { s = d = e - E_POS; }          // self loops
}

// pass 1: score = leaky_relu(al_s[src]+al_d[dst]); segment-max via atomicMax
__global__ void edge_pass1(const int* __restrict__ pei,
                           const float* __restrict__ als,
                           const float* __restrict__ ald,
                           float* __restrict__ scores,
                           unsigned* __restrict__ mkeys) {
  const int e = blockIdx.x * blockDim.x + threadIdx.x;
  if (e >= E_TOT) return;
  int s, d; edge_sd(pei, e, s, d);
  float v = als[s] + ald[d];
  v = (v >= 0.0f) ? v : NEG_SLOPE * v;
  scores[e] = v;
  atomicMax(mkeys + d, enc_f32(v));
}

// pass 2: ex = exp(score - m[dst]); segment-sum via atomicAdd (in-place ex)
__global__ void edge_pass2(const int* __restrict__ pei,
                           const unsigned* __restrict__ mkeys,
                           float* __restrict__ scores,
                           float* __restrict__ ssum) {
  const int e = blockIdx.x * blockDim.x + threadIdx.x;
  if (e >= E_TOT) return;
  int s, d; edge_sd(pei, e, s, d);
  const float ex = expf(scores[e] - dec_f32(mkeys[d]));
  scores[e] = ex;
  atomicAdd(ssum + d, ex);
}

// pass 3: out[dst,:] += h[src,:] * (ex / (s[dst]+eps)); one wave per edge
__global__ void edge_pass3(const int* __restrict__ pei,
                           const float* __restrict__ h,
                           const float* __restrict__ ex,
                           const float* __restrict__ ssum,
                           float* __restrict__ out) {
  const int wave = (blockIdx.x * blockDim.x + threadIdx.x) >> 5;
  const int lane = threadIdx.x & 31;
  if (wave >= E_TOT) return;
  int s, d; edge_sd(pei, wave, s, d);
  const float alpha = ex[wave] / (ssum[d] + EPS_F);
  const float4* hr = (const float4*)(h + (size_t)s * FDIM);
  float* orow = out + (size_t)d * FDIM;
  #pragma unroll
  for (int q = 0; q < 2; ++q) {
    const float4 v = hr[lane + 32 * q];      // coalesced 128b gathers
    const int base = (lane + 32 * q) * 4;
    atomicAdd(orow + base + 0, v.x * alpha);
    atomicAdd(orow + base + 1, v.y * alpha);
    atomicAdd(orow + base + 2, v.z * alpha);
    atomicAdd(orow + base + 3, v.w * alpha);
  }
}

// ---------------------------------------------------------------------------
// logits[e] = h[ei0[e],:] . h[ei1[e],:]  (wave per edge)
// ---------------------------------------------------------------------------
__global__ void logits_kernel(const float* __restrict__ h,
                              const int* __restrict__ ei,
                              float* __restrict__ out) {
  const int wave = (blockIdx.x * blockDim.x + threadIdx.x) >> 5;
  const int lane = threadIdx.x & 31;
  if (wave >= E_PRED) return;
  const int a = ei[wave];
  const int b = ei[E_PRED + wave];
  const float4* ra = (const float4*)(h + (size_t)a * FDIM);
  const float4* rb = (const float4*)(h + (size_t)b * FDIM);
  float acc = 0.0f;
  #pragma unroll
  for (int q = 0; q < 2; ++q) {
    const float4 va = ra[lane + 32 * q];
    const float4 vb = rb[lane + 32 * q];
    acc += va.x * vb.x + va.y * vb.y + va.z * vb.z + va.w * vb.w;
  }
  #pragma unroll
  for (int m = 16; m > 0; m >>= 1) acc += __shfl_xor(acc, m, 32);
  if (lane == 0) out[wave] = acc;
}

// ---------------------------------------------------------------------------
// Host-side orchestration
// ---------------------------------------------------------------------------
static void launch_gat_layer(const float* h_in, const float* W,
                             const float* a_src, const float* a_dst,
                             const float* bias, const int* pei,
                             float* hW, float* out,          // scratch + result
                             float* als, float* ald,
                             unsigned* mkeys, float* ssum, float* scores,
                             hipStream_t stream) {
  const int NF = N_NODES * FDIM;
  // hW = h_in @ W
  wmma_gemm_f32<<<dim3((N_NODES * 16 * 32) / 256), dim3(256), 0, stream>>>(
      h_in, W, nullptr, hW, N_NODES);
  // attention projections
  rowdot2_kernel<<<dim3((N_NODES * 32 + 255) / 256), dim3(256), 0, stream>>>(
      hW, a_src, a_dst, als, ald, N_NODES);
  // init segment buffers + bias-seeded output
  init_ms_kernel<<<dim3((N_NODES + 255) / 256), dim3(256), 0, stream>>>(
      mkeys, ssum, N_NODES);
  init_out_bias<<<dim3((NF + 255) / 256), dim3(256), 0, stream>>>(
      out, bias, NF);
  // segment softmax + aggregation
  edge_pass1<<<dim3((E_TOT + 255) / 256), dim3(256), 0, stream>>>(
      pei, als, ald, scores, mkeys);
  edge_pass2<<<dim3((E_TOT + 255) / 256), dim3(256), 0, stream>>>(
      pei, mkeys, scores, ssum);
  edge_pass3<<<dim3((E_TOT * 32 + 255) / 256), dim3(256), 0, stream>>>(
      pei, hW, scores, ssum, out);
}

extern "C" void kernel_launch(void* const* d_in, const int* in_sizes, int n_in,
                              void* d_out, int out_size, void* d_ws,
                              size_t ws_size, hipStream_t stream) {
  (void)in_sizes; (void)n_in; (void)out_size; (void)ws_size;
  const float* x      = (const float*)d_in[0];
  const int*   pei    = (const int*)  d_in[1];   // [2, 320000]
  const int*   ei     = (const int*)  d_in[2];   // [2, 100000]
  const float* Win_W  = (const float*)d_in[3];
  const float* Win_b  = (const float*)d_in[4];
  const float* W1     = (const float*)d_in[5];
  const float* a1_src = (const float*)d_in[6];
  const float* a1_dst = (const float*)d_in[7];
  const float* b1     = (const float*)d_in[8];
  const float* W2     = (const float*)d_in[9];
  const float* a2_src = (const float*)d_in[10];
  const float* a2_dst = (const float*)d_in[11];
  const float* b2     = (const float*)d_in[12];
  float* logits = (float*)d_out;

  // workspace layout (floats)
  const size_t NF = (size_t)N_NODES * FDIM;      // 2,560,000
  float* ws = (float*)d_ws;
  float*    hA     = ws;                          // [N,F]
  float*    hB     = ws + NF;                     // [N,F]
  float*    hC     = ws + 2 * NF;                 // [N,F]
  float*    als    = ws + 3 * NF;                 // [N]
  float*    ald    = als + N_NODES;               // [N]
  unsigned* mkeys  = (unsigned*)(ald + N_NODES);  // [N]
  float*    ssum   = (float*)mkeys + N_NODES;     // [N]
  float*    scores = ssum + N_NODES;              // [E_TOT]

  // 1) input weighting: hA = x @ Win_W + Win_b
  wmma_gemm_f32<<<dim3((N_NODES * 16 * 32) / 256), dim3(256), 0, stream>>>(
      x, Win_W, Win_b, hA, N_NODES);

  // 2) GAT layer 1: hC = gat(hA; W1, a1, b1)   (hB = hA @ W1 scratch)
  launch_gat_layer(hA, W1, a1_src, a1_dst, b1, pei, hB, hC,
                   als, ald, mkeys, ssum, scores, stream);

  // 3) GAT layer 2: hB = gat(hC; W2, a2, b2)   (hA = hC @ W2 scratch)
  launch_gat_layer(hC, W2, a2_src, a2_dst, b2, pei, hA, hB,
                   als, ald, mkeys, ssum, scores, stream);

  // 4) logits[e] = <hB[ei0[e]], hB[ei1[e]]>
  logits_kernel<<<dim3((E_PRED * 32 + 255) / 256), dim3(256), 0, stream>>>(
      hB, ei, logits);
}